// SparseLinear_12962211299532
// MI455X (gfx1250) — compile-verified
//
#include <hip/hip_runtime.h>
#include <stdint.h>

// Problem constants (match reference).
#define IN_F   50000
#define OUT_F  50000
#define BATCH  128
#define TILE   256   // edges staged per LDS tile (== blockDim)
#define EPB    2048  // edges per block

// ---------------- CDNA5 async-copy helpers ----------------
#if __has_builtin(__builtin_amdgcn_s_wait_asynccnt)
#define WAIT_ASYNC0() __builtin_amdgcn_s_wait_asynccnt(0)
#else
#define WAIT_ASYNC0() asm volatile("s_wait_asynccnt 0x0" ::: "memory")
#endif

// Per-lane async copy: copies 16 bytes from the global address gaddr to the LDS byte address lds_byte_addr.
// gfx1250 GLOBAL_LOAD_ASYNC_TO_LDS_B128, tracked by ASYNCcnt.
__device__ __forceinline__ void async_load_b128_to_lds(uint32_t lds_byte_addr,
                                                       const void* gaddr) {
  asm volatile("global_load_async_to_lds_b128 %0, %1, off"
               :
               : "v"(lds_byte_addr), "v"((uint64_t)(uintptr_t)gaddr)
               : "memory");
}

__device__ __forceinline__ uint32_t lds_addr_of(const void* p) {
  // Low 32 bits of the flat address of an LDS object == wave-relative LDS byte
  // address (LDS aperture mapping).
  return (uint32_t)(uintptr_t)p;
}

// Four hardware fp32 atomics (no-return, STOREcnt-tracked) sharing ONE 64-bit
// address VGPR pair; the 24-bit signed IOFFSET immediate addresses the rest.
__device__ __forceinline__ void gatomic_add_f32x4_contig(float* base,
                                                         float a, float b,
                                                         float c, float d) {
  asm volatile("global_atomic_add_f32 %0, %1, off\n\t"
               "global_atomic_add_f32 %0, %2, off offset:4\n\t"
               "global_atomic_add_f32 %0, %3, off offset:8\n\t"
               "global_atomic_add_f32 %0, %4, off offset:12"
               :
               : "v"((uint64_t)(uintptr_t)base), "v"(a), "v"(b), "v"(c), "v"(d)
               : "memory");
}

// Two fp32 atomics off one base; STRIDE must fit signed 24-bit IOFFSET
// (|STRIDE| <= 8388607).
__device__ __forceinline__ void gatomic_add_f32x2_stride(float* base, int stride_bytes_unused,
                                                         float a, float b) {
  (void)stride_bytes_unused;
}
#define GATOMIC_ADD_F32X2_STRIDE(baseptr, STRIDE_BYTES, a, b)                  \
  asm volatile("global_atomic_add_f32 %0, %1, off\n\t"                         \
               "global_atomic_add_f32 %0, %2, off offset:%c3"                  \
               :                                                               \
               : "v"((uint64_t)(uintptr_t)(baseptr)), "v"(a), "v"(b),          \
                 "i"(STRIDE_BYTES)                                             \
               : "memory")

// ---------------- x transpose: x[128][IN] -> xT[IN][128] ----------------
__global__ __launch_bounds__(256) void transpose_x_kernel(
    const float* __restrict__ x, float* __restrict__ xT) {
  __shared__ float tile[32][33];
  const int c0 = blockIdx.x * 32;
  const int b0 = blockIdx.y * 32;
  const int tx = threadIdx.x, ty = threadIdx.y;
#pragma unroll
  for (int j = 0; j < 4; ++j) {
    const int b = b0 + ty + j * 8;
    const int c = c0 + tx;
    if (c < IN_F) tile[ty + j * 8][tx] = x[(size_t)b * IN_F + c];
  }
  __syncthreads();
#pragma unroll
  for (int j = 0; j < 4; ++j) {
    const int c = c0 + ty + j * 8;
    const int b = b0 + tx;
    if (c < IN_F) xT[(size_t)c * BATCH + b] = tile[tx][ty + j * 8];
  }
}

// ------------- edge tile staging (double-buffered async pipeline) -------------
// Full tile (256 edges): threads 0-63 stage rows, 64-127 cols, 128-191 vals,
// one b128 (4 edges) each. Partial tail tile: plain per-thread loads.
__device__ __forceinline__ void stage_tile(const int* __restrict__ rows,
                                           const int* __restrict__ cols,
                                           const float* __restrict__ vals,
                                           int base, int nnz,
                                           int* s_rows, int* s_cols, float* s_vals) {
  const int tid = threadIdx.x;
  if (base + TILE <= nnz) {
    const int group = tid >> 6;   // 0..3
    const int sub   = tid & 63;   // 4 edges per thread
    if (group == 0) {
      async_load_b128_to_lds(lds_addr_of(&s_rows[sub * 4]), rows + base + sub * 4);
    } else if (group == 1) {
      async_load_b128_to_lds(lds_addr_of(&s_cols[sub * 4]), cols + base + sub * 4);
    } else if (group == 2) {
      async_load_b128_to_lds(lds_addr_of(&s_vals[sub * 4]), vals + base + sub * 4);
    }
  } else {
    const int idx = base + tid;
    if (idx < nnz) {
      s_rows[tid] = rows[idx];
      s_cols[tid] = cols[idx];
      s_vals[tid] = vals[idx];
    }
  }
}

// ------------- edge scatter (transposed layout, coalesced atomics) -------------
__global__ __launch_bounds__(256) void scatter_t_kernel(
    const int* __restrict__ rows, const int* __restrict__ cols,
    const float* __restrict__ vals, const float4* __restrict__ xT4,
    float* __restrict__ yT, int nnz) {
  __shared__ __align__(16) int   s_rows[2][TILE];
  __shared__ __align__(16) int   s_cols[2][TILE];
  __shared__ __align__(16) float s_vals[2][TILE];

  const int lane = threadIdx.x & 31;
  const int wave = threadIdx.x >> 5;  // 0..7
  const int base = blockIdx.x * EPB;
  if (base >= nnz) return;
  const int end = min(base + EPB, nnz);

  int cur = 0;
  stage_tile(rows, cols, vals, base, nnz, s_rows[0], s_cols[0], s_vals[0]);
  WAIT_ASYNC0();
  __syncthreads();

  for (int t = base; t < end; t += TILE) {
    const int nxt = t + TILE;
    if (nxt < end) {
      stage_tile(rows, cols, vals, nxt, nnz,
                 s_rows[cur ^ 1], s_cols[cur ^ 1], s_vals[cur ^ 1]);
    }
    const int nv = min(TILE, end - t);
    // One edge per wave per step: scalar row/col/val broadcast from LDS,
    // one 512B-coalesced float4 gather, 4 coalesced fp32 atomics (512B span)
    // issued off a single address pair via IOFFSET immediates.
    for (int e = wave; e < nv; e += 8) {
      const int   row = s_rows[cur][e];
      const int   col = s_cols[cur][e];
      const float v   = s_vals[cur][e];
      const float4 xv = xT4[(size_t)col * (BATCH / 4) + lane];
      float* dst = yT + (size_t)row * BATCH + lane * 4;
      gatomic_add_f32x4_contig(dst, v * xv.x, v * xv.y, v * xv.z, v * xv.w);
    }
    WAIT_ASYNC0();   // next tile's async copies have landed
    __syncthreads(); // all waves done with buffer `cur`, next data visible
    cur ^= 1;
  }
}

// ------------- finalize: y[b][o] = yT[o][b] + bias[o] -------------
__global__ __launch_bounds__(256) void finalize_kernel(
    const float* __restrict__ yT, const float* __restrict__ bias,
    float* __restrict__ y) {
  __shared__ float tile[32][33];
  const int o0 = blockIdx.x * 32;
  const int b0 = blockIdx.y * 32;
  const int tx = threadIdx.x, ty = threadIdx.y;
#pragma unroll
  for (int j = 0; j < 4; ++j) {
    const int o = o0 + ty + j * 8;
    if (o < OUT_F) tile[ty + j * 8][tx] = yT[(size_t)o * BATCH + b0 + tx];
  }
  __syncthreads();
#pragma unroll
  for (int j = 0; j < 4; ++j) {
    const int o = o0 + tx;
    const int b = b0 + ty + j * 8;
    if (o < OUT_F) y[(size_t)b * OUT_F + o] = tile[tx][ty + j * 8] + bias[o];
  }
}

// ---------------- fallback path (no workspace): direct layout ----------------
__global__ __launch_bounds__(256) void init_direct_kernel(
    const float* __restrict__ bias, float* __restrict__ y) {
  const int o = blockIdx.x * 256 + threadIdx.x;
  const int b = blockIdx.y;
  if (o < OUT_F) y[(size_t)b * OUT_F + o] = bias[o];
}

__global__ __launch_bounds__(256) void scatter_direct_kernel(
    const int* __restrict__ rows, const int* __restrict__ cols,
    const float* __restrict__ vals, const float* __restrict__ x,
    float* __restrict__ y, int nnz) {
  __shared__ __align__(16) int   s_rows[2][TILE];
  __shared__ __align__(16) int   s_cols[2][TILE];
  __shared__ __align__(16) float s_vals[2][TILE];

  const int lane = threadIdx.x & 31;
  const int wave = threadIdx.x >> 5;
  const int base = blockIdx.x * EPB;
  if (base >= nnz) return;
  const int end = min(base + EPB, nnz);

  int cur = 0;
  stage_tile(rows, cols, vals, base, nnz, s_rows[0], s_cols[0], s_vals[0]);
  WAIT_ASYNC0();
  __syncthreads();

  for (int t = base; t < end; t += TILE) {
    const int nxt = t + TILE;
    if (nxt < end) {
      stage_tile(rows, cols, vals, nxt, nnz,
                 s_rows[cur ^ 1], s_cols[cur ^ 1], s_vals[cur ^ 1]);
    }
    const int nv = min(TILE, end - t);
    for (int e = wave; e < nv; e += 8) {
      const int   row = s_rows[cur][e];
      const int   col = s_cols[cur][e];
      const float v   = s_vals[cur][e];
      // Batches lane, lane+32, lane+64, lane+96. Row stride between pairs is
      // 32*OUT_F*4 = 6,400,000 B which fits the signed 24-bit IOFFSET
      // (+/- 8,388,607); the 64-batch span does not, so use two bases.
      const float* xs  = x + (size_t)lane * IN_F + col;
      float*       yd0 = y + (size_t)lane * OUT_F + row;
      float*       yd1 = yd0 + (size_t)64 * OUT_F;
      const float x0 = xs[0];
      const float x1 = xs[(size_t)32 * IN_F];
      const float x2 = xs[(size_t)64 * IN_F];
      const float x3 = xs[(size_t)96 * IN_F];
      GATOMIC_ADD_F32X2_STRIDE(yd0, 6400000, v * x0, v * x1);
      GATOMIC_ADD_F32X2_STRIDE(yd1, 6400000, v * x2, v * x3);
    }
    WAIT_ASYNC0();
    __syncthreads();
    cur ^= 1;
  }
}

// ---------------- host launcher ----------------
extern "C" void kernel_launch(void* const* d_in, const int* in_sizes, int n_in,
                              void* d_out, int out_size, void* d_ws, size_t ws_size,
                              hipStream_t stream) {
  const float* x    = (const float*)d_in[0];
  const int*   rows = (const int*)d_in[1];
  const int*   cols = (const int*)d_in[2];
  const float* vals = (const float*)d_in[3];
  const float* bias = (const float*)d_in[4];
  float*       y    = (float*)d_out;
  const int    nnz  = in_sizes[1];

  const size_t xT_bytes = (size_t)IN_F * BATCH * sizeof(float);
  const size_t yT_bytes = (size_t)OUT_F * BATCH * sizeof(float);
  const int scatter_blocks = (nnz + EPB - 1) / EPB;

  if (ws_size >= xT_bytes + yT_bytes) {
    float* xT = (float*)d_ws;
    float* yT = (float*)((char*)d_ws + xT_bytes);

    (void)hipMemsetAsync(yT, 0, yT_bytes, stream);

    dim3 tb(32, 8);
    transpose_x_kernel<<<dim3((IN_F + 31) / 32, BATCH / 32), tb, 0, stream>>>(x, xT);
    scatter_t_kernel<<<scatter_blocks, 256, 0, stream>>>(
        rows, cols, vals, (const float4*)xT, yT, nnz);
    finalize_kernel<<<dim3((OUT_F + 31) / 32, BATCH / 32), tb, 0, stream>>>(yT, bias, y);
  } else {
    init_direct_kernel<<<dim3((OUT_F + 255) / 256, BATCH), 256, 0, stream>>>(bias, y);
    scatter_direct_kernel<<<scatter_blocks, 256, 0, stream>>>(rows, cols, vals, x, y, nnz);
  }
}